// PositionRankingLoss_79594333929900
// MI455X (gfx1250) — compile-verified
//
#include <hip/hip_runtime.h>

// PositionRankingLoss on MI455X (gfx1250, wave32).
//
// total = sum_{t[i]<t[j]} max(1 - (p[j]-p[i]), 0);  out = total/count (scalar).
// VALU-bound O(n^2) problem (67M pairs, 64KB data): LDS-tiled brute force,
// exact int counting, fp32 tree sums. v_wmma_f32_16x16x4_f32 (A = ones) is
// used as an exact one-instruction 32->16 wave reduction of fp32 partials.

#define BLOCK  256   // 8 wave32s
#define JCHUNK 512   // j-tile per block (4KB LDS: 2KB p + 2KB t)

typedef __attribute__((ext_vector_type(2))) float v2f;
typedef __attribute__((ext_vector_type(8))) float v8f;

// Exact fp32 wave(32) sum. WMMA path: D = Ones(16x4) x B(4x16) gives column
// sums of B; placing each lane's value in B-VGPR0 (B-VGPR1 = 0) means the 16
// column sums collectively contain all 32 lane values exactly once, whatever
// B's internal (k,n)->lane mapping is. D layout is documented: D-VGPR0 lane L
// holds colsum[L%16] (rows of D identical since A's rows are identical).
// Four xor-shuffles finish 16 -> 1.
__device__ __forceinline__ float wave_sum_f32(float s) {
#if defined(__gfx1250__) && __has_builtin(__builtin_amdgcn_wmma_f32_16x16x4_f32)
  v2f a; a[0] = 1.0f; a[1] = 1.0f;      // all-ones A (layout invariant)
  v2f b; b[0] = s;    b[1] = 0.0f;      // lane partial into B
  v8f c = {};
  v8f d = __builtin_amdgcn_wmma_f32_16x16x4_f32(
      /*neg_a=*/false, a, /*neg_b=*/false, b,
      /*c_mod=*/(short)0, c, /*reuse_a=*/false, /*reuse_b=*/false);
  float v = d[0];                        // lane L: colsum[L % 16]
  v += __shfl_xor(v, 1, 32);
  v += __shfl_xor(v, 2, 32);
  v += __shfl_xor(v, 4, 32);
  v += __shfl_xor(v, 8, 32);
  return v;                              // every lane: wave total
#else
  for (int m = 16; m >= 1; m >>= 1) s += __shfl_xor(s, m, 32);
  return s;
#endif
}

__device__ __forceinline__ int wave_sum_i32(int c) {
  for (int m = 16; m >= 1; m >>= 1) c += __shfl_xor(c, m, 32);
  return c;
}

__global__ void __launch_bounds__(BLOCK)
rank_hinge_partials(const float* __restrict__ pred,
                    const int* __restrict__ target,
                    float* __restrict__ psum,
                    int* __restrict__ pcnt,
                    int n, int nJ) {
  __shared__ float sp[JCHUNK];
  __shared__ int   st[JCHUNK];
  __shared__ float wsum[BLOCK / 32];
  __shared__ int   wcnt[BLOCK / 32];

  const int tid   = threadIdx.x;
  const int rb    = (int)blockIdx.x / nJ;   // row block
  const int jc    = (int)blockIdx.x % nJ;   // j chunk
  const int jbase = jc * JCHUNK;
  const int jlen  = min(JCHUNK, n - jbase);

  // Cooperative chunk load; pad tail with t = INT_MIN so (ti < tj) is false.
  for (int k = tid; k < JCHUNK; k += BLOCK) {
    if (k < jlen) {
      sp[k] = pred[jbase + k];
      st[k] = target[jbase + k];
    } else {
      sp[k] = 0.0f;
      st[k] = (int)0x80000000;
    }
  }
  __syncthreads();

  const int i = rb * BLOCK + tid;
  float s  = 0.0f;
  int   cn = 0;
  if (i < n) {
    const float pi = pred[i];
    const int   ti = target[i];
    // Uniform j across the wave -> LDS broadcast reads, no bank conflicts.
#pragma unroll 8
    for (int j = 0; j < JCHUNK; ++j) {
      const int   tj   = st[j];
      const float term = fmaxf(1.0f - (sp[j] - pi), 0.0f);
      const bool  lt   = ti < tj;
      s  += lt ? term : 0.0f;
      cn += lt ? 1 : 0;
    }
  }

  // Full EXEC here (no early returns) -> WMMA reduction is legal.
  const float ws = wave_sum_f32(s);
  const int   wc = wave_sum_i32(cn);

  const int lane = tid & 31, wave = tid >> 5;
  if (lane == 0) { wsum[wave] = ws; wcnt[wave] = wc; }
  __syncthreads();

  if (tid == 0) {
    float bs = 0.0f; int bc = 0;
#pragma unroll
    for (int w = 0; w < BLOCK / 32; ++w) { bs += wsum[w]; bc += wcnt[w]; }
    psum[blockIdx.x] = bs;
    pcnt[blockIdx.x] = bc;   // exact int count per block
  }
}

__global__ void __launch_bounds__(BLOCK)
rank_hinge_final(const float* __restrict__ psum,
                 const int* __restrict__ pcnt,
                 float* __restrict__ out, int P) {
  __shared__ float wsum[BLOCK / 32];
  __shared__ int   wcnt[BLOCK / 32];

  float s  = 0.0f;
  int   cn = 0;
  for (int k = threadIdx.x; k < P; k += BLOCK) {
    s  += psum[k];
    cn += pcnt[k];
  }

  const float ws = wave_sum_f32(s);
  const int   wc = wave_sum_i32(cn);

  const int lane = threadIdx.x & 31, wave = threadIdx.x >> 5;
  if (lane == 0) { wsum[wave] = ws; wcnt[wave] = wc; }
  __syncthreads();

  if (threadIdx.x == 0) {
    float total = 0.0f; int count = 0;
#pragma unroll
    for (int w = 0; w < BLOCK / 32; ++w) { total += wsum[w]; count += wcnt[w]; }
    out[0] = (count > 0) ? (total / (float)count) : 0.0f;  // single int->f32 cast
  }
}

extern "C" void kernel_launch(void* const* d_in, const int* in_sizes, int n_in,
                              void* d_out, int out_size, void* d_ws, size_t ws_size,
                              hipStream_t stream) {
  (void)n_in; (void)out_size; (void)ws_size;
  const float* pred   = (const float*)d_in[0];
  const int*   target = (const int*)d_in[1];
  const int n = in_sizes[0];

  const int nRB = (n + BLOCK - 1) / BLOCK;    // row blocks (32 for n=8192)
  const int nJ  = (n + JCHUNK - 1) / JCHUNK;  // j chunks   (16 for n=8192)
  const int P   = nRB * nJ;                   // 512 partials -> 4KB of d_ws

  float* psum = (float*)d_ws;
  int*   pcnt = (int*)((char*)d_ws + (size_t)P * sizeof(float));

  rank_hinge_partials<<<P, BLOCK, 0, stream>>>(pred, target, psum, pcnt, n, nJ);
  rank_hinge_final<<<1, BLOCK, 0, stream>>>(psum, pcnt, (float*)d_out, P);
}